// LightGCN_69234872812360
// MI455X (gfx1250) — compile-verified
//
#include <hip/hip_runtime.h>

#define N_USERS 100000
#define N_ITEMS 50000
#define NNODES  (N_USERS + N_ITEMS)
#define DIM     64
#define NEDGES  3200000
#define NLAYERS 3

typedef float v2f __attribute__((ext_vector_type(2)));
typedef float v8f __attribute__((ext_vector_type(8)));

// ---------------------------------------------------------------------------
// emb0 = concat(user_emb, item_emb); acc = emb0
// ---------------------------------------------------------------------------
__global__ void lg_concat_init(const float* __restrict__ ue,
                               const float* __restrict__ ie,
                               float* __restrict__ emb,
                               float* __restrict__ acc) {
  int i = blockIdx.x * blockDim.x + threadIdx.x;      // float4 index
  const int totalV4 = NNODES * (DIM / 4);
  if (i >= totalV4) return;
  const int userV4 = N_USERS * (DIM / 4);
  float4 v = (i < userV4) ? ((const float4*)ue)[i]
                          : ((const float4*)ie)[i - userV4];
  ((float4*)emb)[i] = v;
  ((float4*)acc)[i] = v;
}

// ---------------------------------------------------------------------------
// One LightGCN layer: Y[dst] += vals * X[src], via WMMA row-merged SpMM.
// Each wave32 processes a block of 16 edges:
//   P (16x16): P[r][e] = vals[e] if dst[e]==dst[r] and r is the leader
//              (first edge in block with that dst), else 0
//   D (16x64) = P x Xsrc (16x64)   -- 16 chained V_WMMA_F32_16X16X4_F32
// Non-leader D rows are exactly zero, so the scatter is UNCONDITIONAL
// (adding 0.0 is harmless) -> no exec-mask branching, pure atomic clauses.
// ---------------------------------------------------------------------------
__global__ void lg_spmm_wmma(const float* __restrict__ X,
                             float* __restrict__ Y,
                             const int* __restrict__ src,
                             const int* __restrict__ dst,
                             const float* __restrict__ vals) {
  const int lane   = threadIdx.x & 31;
  const int r      = lane & 15;        // row (lanes 16-31 mirror 0-15)
  const int kHi    = lane >> 4;        // 0: K=0,1  1: K=2,3  (A layout)
  const int wave   = (blockIdx.x * blockDim.x + threadIdx.x) >> 5;
  const int nWaves = (gridDim.x * blockDim.x) >> 5;
  const int nBlk   = NEDGES / 16;      // 200000, exact

  for (int blk = wave; blk < nBlk; blk += nWaves) {
    const int e0 = blk * 16;

    if (blk + nWaves < nBlk) {         // global_prefetch_b8 next block's meta
      __builtin_prefetch(&src[(blk + nWaves) * 16], 0, 1);
      __builtin_prefetch(&dst[(blk + nWaves) * 16], 0, 1);
      __builtin_prefetch(&vals[(blk + nWaves) * 16], 0, 1);
    }

    // per-lane copy of edge r's metadata (both wave halves hold r = lane&15)
    const int   dL = dst[e0 + r];
    const int   sL = src[e0 + r];
    const float vL = vals[e0 + r];
    const int   myDst = dL;

    // one broadcast of all 16 dst values; feeds eq flags, leader, scatter
    int dB[16];
#pragma unroll
    for (int e = 0; e < 16; ++e) dB[e] = __shfl(dL, e, 32);

    bool eq[16];
#pragma unroll
    for (int e = 0; e < 16; ++e) eq[e] = (dB[e] == myDst);

    // leader: no earlier edge in the block has the same dst
    int leader = 1;
#pragma unroll
    for (int rp = 0; rp < 15; ++rp)
      if (rp < r && eq[rp]) leader = 0;

    // A-matrix chunks (16x4 f32, 2 VGPRs each), independent of dim-chunk
    v2f aArr[4];
    int sB0[4], sB1[4];
#pragma unroll
    for (int kc = 0; kc < 4; ++kc) {
      const bool e0q = kHi ? eq[4 * kc + 2] : eq[4 * kc];
      const bool e1q = kHi ? eq[4 * kc + 3] : eq[4 * kc + 1];
      const float v0 = __shfl(vL, 4 * kc + 2 * kHi,     32);
      const float v1 = __shfl(vL, 4 * kc + 2 * kHi + 1, 32);
      aArr[kc].x = (leader && e0q) ? v0 : 0.0f;
      aArr[kc].y = (leader && e1q) ? v1 : 0.0f;
      // B-matrix source rows (4x16 f32: v0 = rows K0/K1, v1 = rows K2/K3)
      sB0[kc] = __shfl(sL, 4 * kc + kHi,     32);
      sB1[kc] = __shfl(sL, 4 * kc + 2 + kHi, 32);
    }

    // scatter rows: VGPR g holds M = g + 8*kHi; dst comes from dB (no shfl)
    int dM[8];
#pragma unroll
    for (int g = 0; g < 8; ++g) dM[g] = kHi ? dB[g + 8] : dB[g];

    // 4 chunks of 16 output dims; K=16 via 4 chained 16x16x4 WMMAs
#pragma unroll
    for (int nc = 0; nc < 4; ++nc) {
      v8f c = {0.f, 0.f, 0.f, 0.f, 0.f, 0.f, 0.f, 0.f};
#pragma unroll
      for (int kc = 0; kc < 4; ++kc) {
        v2f b;
        b.x = X[sB0[kc] * DIM + nc * 16 + r];
        b.y = X[sB1[kc] * DIM + nc * 16 + r];
        c = __builtin_amdgcn_wmma_f32_16x16x4_f32(
                false, aArr[kc], false, b, (short)0, c, false, false);
      }
#pragma unroll
      for (int g = 0; g < 8; ++g) {
        atomicAdd(&Y[dM[g] * DIM + nc * 16 + r], c[g]);   // unconditional
      }
    }
  }
}

// ---------------------------------------------------------------------------
// acc += e   (float4)
// ---------------------------------------------------------------------------
__global__ void lg_axpy_acc(float* __restrict__ acc, const float* __restrict__ e) {
  int i = blockIdx.x * blockDim.x + threadIdx.x;
  if (i >= NNODES * (DIM / 4)) return;
  float4 a = ((const float4*)acc)[i];
  float4 b = ((const float4*)e)[i];
  a.x += b.x; a.y += b.y; a.z += b.z; a.w += b.w;
  ((float4*)acc)[i] = a;
}

// ---------------------------------------------------------------------------
// acc *= 1/(NLAYERS+1)
// ---------------------------------------------------------------------------
__global__ void lg_scale(float* __restrict__ acc) {
  int i = blockIdx.x * blockDim.x + threadIdx.x;
  if (i >= NNODES * (DIM / 4)) return;
  const float s = 1.0f / (float)(NLAYERS + 1);
  float4 a = ((float4*)acc)[i];
  a.x *= s; a.y *= s; a.z *= s; a.w *= s;
  ((float4*)acc)[i] = a;
}

extern "C" void kernel_launch(void* const* d_in, const int* in_sizes, int n_in,
                              void* d_out, int out_size, void* d_ws, size_t ws_size,
                              hipStream_t stream) {
  const float* ue   = (const float*)d_in[0];
  const float* ie   = (const float*)d_in[1];
  const int*   src  = (const int*)d_in[2];
  const int*   dst  = (const int*)d_in[3];
  const float* vals = (const float*)d_in[4];

  float* acc = (float*)d_out;                       // running sum of layers
  float* A   = (float*)d_ws;                        // emb_prev
  float* B   = A + (size_t)NNODES * DIM;            // emb_next
  const size_t embBytes = (size_t)NNODES * DIM * sizeof(float);

  const int v4 = NNODES * DIM / 4;
  dim3 eb(256), eg((v4 + 255) / 256);

  lg_concat_init<<<eg, eb, 0, stream>>>(ue, ie, A, acc);

  for (int l = 0; l < NLAYERS; ++l) {
    hipMemsetAsync(B, 0, embBytes, stream);
    lg_spmm_wmma<<<4096, 256, 0, stream>>>(A, B, src, dst, vals);
    lg_axpy_acc<<<eg, eb, 0, stream>>>(acc, B);
    float* t = A; A = B; B = t;                     // ping-pong
  }

  lg_scale<<<eg, eb, 0, stream>>>(acc);
}